// JDEBBoxPostProcess_78847009620436
// MI455X (gfx1250) — compile-verified
//
#include <hip/hip_runtime.h>
#include <stdint.h>

// ---------------- problem constants (from reference) ----------------
#define N_CANDS   4000000
#define M_SEL     3000
#define MP        3072          // M padded to multiple of 16 (192 tiles)
#define NSORT     4096          // power-of-two for bitonic sort
#define NT        192           // 16-wide tiles per dimension
#define NW32      96            // 32-bit words per mask row (MP/32)
#define MAX_DETS  300
#define NMS_T     0.6f
#define SCORE_T   0.01f

// ---------------- workspace layout (float indices unless noted) -----
// [0, 4*MP)   : AoS boxes float4 {x1,y1,x2,y2}[MP]
// [4*MP,5*MP) : AREA[MP]          (boxes+area = 61440 B, TDM-copied)
#define WAR   (4*MP)
#define WS_   (5*MP)            // sorted scores
#define WORD_ (6*MP)            // int: order[] (sorted pos -> original selection pos)
#define WNV   (7*MP)            // int: count of scores > SCORE_T
#define MASK_BYTE_OFF (128*1024)  // uint16 mask16[MP][NT] = 1,179,648 B

typedef __attribute__((ext_vector_type(2))) float    v2f;
typedef __attribute__((ext_vector_type(4))) float    v4f;
typedef __attribute__((ext_vector_type(8))) float    v8f;
typedef __attribute__((ext_vector_type(3))) unsigned v3u;
typedef __attribute__((ext_vector_type(4))) unsigned v4u;
typedef __attribute__((ext_vector_type(4))) int      v4i;
typedef __attribute__((ext_vector_type(8))) int      v8i;

// ==================================================================
// Kernel A: gather 3000 scores, bitonic sort descending (key=score,
// payload=selection position), emit sorted AoS boxes + areas + order.
// ==================================================================
__global__ __launch_bounds__(1024)
void nms_gather_sort(const float* __restrict__ yolo, const int* __restrict__ bidx,
                     float* __restrict__ ws) {
  __shared__ float skey[NSORT];
  __shared__ int   sidx[NSORT];
  __shared__ int   lcnt;
  const int tid = threadIdx.x;
  if (tid == 0) lcnt = 0;

  for (int k = tid; k < NSORT; k += 1024) {
    float s = -1e30f;
    if (k < M_SEL) {
      int ci = bidx[k];
      s = yolo[(size_t)ci * 5 + 4];
    }
    skey[k] = s;
    sidx[k] = k;
  }
  __syncthreads();

  // bitonic sort, descending
  for (int kk = 2; kk <= NSORT; kk <<= 1) {
    for (int j = kk >> 1; j > 0; j >>= 1) {
      for (int t = tid; t < NSORT; t += 1024) {
        int ixj = t ^ j;
        if (ixj > t) {
          float a = skey[t], b = skey[ixj];
          bool up = ((t & kk) == 0);
          bool sw = up ? (a < b) : (a > b);   // descending overall
          if (sw) {
            skey[t] = b; skey[ixj] = a;
            int tv = sidx[t]; sidx[t] = sidx[ixj]; sidx[ixj] = tv;
          }
        }
      }
      __syncthreads();
    }
  }

  for (int k = tid; k < MP; k += 1024) {
    v4f box = { -2e8f, -2e8f, -2e8f, -2e8f };
    float s = -1.0f;
    int od = -1;
    if (k < M_SEL) {
      od = sidx[k];
      s  = skey[k];
      const float* r = yolo + (size_t)bidx[od] * 5;
      box.x = r[0]; box.y = r[1]; box.z = r[2]; box.w = r[3];
      if (s > SCORE_T) atomicAdd(&lcnt, 1);
    }
    *(v4f*)(ws + 4 * (size_t)k) = box;                          // one b128 store
    ws[WAR + k] = (box.z - box.x + 1.0f) * (box.w - box.y + 1.0f);
    ws[WS_ + k] = s;
    ((int*)ws)[WORD_ + k] = od;
  }
  __syncthreads();
  if (tid == 0) ((int*)ws)[WNV] = lcnt;
}

// ==================================================================
// Kernel B: upper-triangular suppression bitmask. 96 waves; each wave
// handles the balanced row-tile pair {w, 191-w}. Diagonal tiles use
// compile-time bit masks (no integer compares). Box AoS staged via
// the Tensor Data Mover; T*(areaSum) via WMMA f32 16x16x4 with the
// 0.6 scale folded into the B operand; ballots redistributed via a
// branchless cndmask select tree; packed b32 stores.
// ==================================================================
__global__ __launch_bounds__(128)
void nms_mask(const float* __restrict__ ws, unsigned short* __restrict__ mask16) {
  __shared__ __align__(16) float lbox[5 * MP];   // box4[MP] | AREA[MP] (61440 B)
  const int tid  = threadIdx.x;
  const int lane = tid & 31;

#if __has_builtin(__builtin_amdgcn_tensor_load_to_lds)
  // --- TDM: one 1D DMA of 15360 dwords, issued by wave 0 of the block ---
  if (tid < 32) {
    unsigned lds_addr = (unsigned)(uintptr_t)(&lbox[0]);
    unsigned long long ga = (unsigned long long)(uintptr_t)ws;
    const unsigned ELEMS = 5u * MP;          // 15360 dwords
    v4u g0; v8i g1; v4i gz4 = {0, 0, 0, 0};
    g0[0] = 1u;                               // count=1, user descriptor
    g0[1] = lds_addr;                         // lds_addr
    g0[2] = (unsigned)(ga & 0xffffffffu);     // global_addr[31:0]
    g0[3] = (unsigned)((ga >> 32) & 0x01ffffffu) | (2u << 30);  // addr[56:32] | type=2
    g1[0] = (int)(2u << 16);                  // data_size=4B, no mask/iter/pad
    g1[1] = (int)((ELEMS & 0xffffu) << 16);   // tensor_dim0[15:0]
    g1[2] = (int)((ELEMS >> 16) | (1u << 16));          // tensor_dim0[31:16] | tensor_dim1=1
    g1[3] = (int)((ELEMS & 0xffffu) << 16);   // tile_dim0 = 15360
    g1[4] = 0;                                // tile_dim1=0 (unused), tile_dim2=0
    g1[5] = (int)ELEMS;                       // tensor_dim0_stride[31:0]
    g1[6] = 0;
    g1[7] = 0;
#if __clang_major__ >= 23
    v8i gz8 = {0, 0, 0, 0, 0, 0, 0, 0};
    __builtin_amdgcn_tensor_load_to_lds(g0, g1, gz4, gz4, gz8, 0);
#else
    __builtin_amdgcn_tensor_load_to_lds(g0, g1, gz4, gz4, 0);
#endif
    __builtin_amdgcn_s_wait_tensorcnt((short)0);
  }
#else
  for (int k = tid; k < 5 * MP; k += 128) lbox[k] = ws[k];
#endif
  __syncthreads();

  const v4f*   box4 = (const v4f*)lbox;
  const float* lar  = lbox + 4 * MP;

  const int wv   = (int)(blockIdx.x * 128 + tid) >> 5;   // 0..95
  const int l15  = lane & 31 & 15;
  const int lb0 = lane & 1, lb1 = lane & 2, lb2 = lane & 4, lb3 = lane & 8;
  const bool low16 = (lane < 16);

#pragma unroll 1
  for (int t = 0; t < 2; ++t) {
    const int ti = (t == 0) ? wv : (NT - 1 - wv);  // balanced pair {w, 191-w}
    const int i0 = ti * 16;
    const int half = (lane >= 16) ? 8 : 0;

    float rx1[8], ry1[8], rx2[8], ry2[8];
#pragma unroll
    for (int v = 0; v < 8; ++v) {
      v4f rb = box4[i0 + half + v];              // one ds b128 per row
      rx1[v] = rb.x; ry1[v] = rb.y; rx2[v] = rb.z; ry2[v] = rb.w;
    }

#if __has_builtin(__builtin_amdgcn_wmma_f32_16x16x4_f32)
    // A(16x4): col0 = area_row, col1 = ones (lanes 0-15 hold K=0,1).
    const float arR = lar[i0 + l15];             // uniform-exec LDS load
    v2f Aop;
    Aop.x = low16 ? arR : 0.0f;
    Aop.y = low16 ? 1.0f : 0.0f;
#else
    float arRow[8];
#pragma unroll
    for (int v = 0; v < 8; ++v) arRow[v] = lar[i0 + half + v];
#endif

    // -------- one 16x16 tile -> packed halfword for lane r<16 --------
    auto tileBits = [&](int tjj, bool diag) -> unsigned {
      const int j0 = tjj * 16;
      const int jc = j0 + l15;
      const v4f cb = box4[jc];                   // one ds b128 for the column

      v8f S;   // S[v] = T * (area_row + area_col)  (scale folded into B)
#if __has_builtin(__builtin_amdgcn_wmma_f32_16x16x4_f32)
      const float arC = lar[j0 + l15];           // uniform-exec LDS load
      v2f Bop;
      Bop.x = low16 ? NMS_T : 0.0f;              // B row0 = T
      Bop.y = low16 ? (NMS_T * arC) : 0.0f;      // B row1 = T * area_col
      v8f C = {};
      S = __builtin_amdgcn_wmma_f32_16x16x4_f32(false, Aop, false, Bop,
                                                (short)0, C, false, false);
#else
      const float arC = lar[jc];
#pragma unroll
      for (int v = 0; v < 8; ++v) S[v] = NMS_T * (arRow[v] + arC);
#endif

      unsigned bl[8];
#pragma unroll
      for (int v = 0; v < 8; ++v) {
        float xx1 = fmaxf(rx1[v], cb.x), yy1 = fmaxf(ry1[v], cb.y);
        float xx2 = fminf(rx2[v], cb.z), yy2 = fminf(ry2[v], cb.w);
        float w = fmaxf(xx2 - xx1 + 1.0f, 0.0f);
        float h = fmaxf(yy2 - yy1 + 1.0f, 0.0f);
        float inter = w * h;
        // iou > T  <=>  inter*(1+T) > T*areaSum   (all terms positive)
        bool p = (inter * (1.0f + NMS_T) > S[v]);
        bl[v] = (unsigned)__ballot(p);
        if (diag) {
          // j > i on the diagonal tile is lane-geometry only:
          //   low16 row v: keep cols > v; high16 row v+8: keep cols > v+8
          const unsigned dmask = ((0xFFFFu << (v + 1)) & 0xFFFFu)
                               | (((0xFFFFu << (v + 9)) & 0xFFFFu) << 16);
          bl[v] &= dmask;
        }
      }
      // branchless redistribution: lane r<16 takes
      //   r<8 ? bl[r]&0xffff : bl[r-8]>>16   via lane-bit select tree
      unsigned s0a = lb0 ? bl[1] : bl[0];
      unsigned s0b = lb0 ? bl[3] : bl[2];
      unsigned s0c = lb0 ? bl[5] : bl[4];
      unsigned s0d = lb0 ? bl[7] : bl[6];
      unsigned s1a = lb1 ? s0b : s0a;
      unsigned s1b = lb1 ? s0d : s0c;
      unsigned g   = lb2 ? s1b : s1a;
      return lb3 ? (g >> 16) : (g & 0xffffu);
    };

    auto storePair = [&](int tj, unsigned h0, unsigned h1) {
      if (low16) {
        unsigned d = h0 | (h1 << 16);
        *(unsigned*)((char*)mask16 + ((size_t)(i0 + lane) * NT + tj) * 2) = d;
      }
    };

    // -------- peeled diagonal pair --------
    const int tjStart = ti & ~1;
    if (ti & 1) storePair(tjStart, 0u, tileBits(tjStart + 1, true));
    else        storePair(tjStart, tileBits(tjStart, true), tileBits(tjStart + 1, false));

    // -------- steady state: strictly above diagonal, pure float compares --------
    for (int tj = tjStart + 2; tj < NT; tj += 2)
      storePair(tj, tileBits(tj, false), tileBits(tj + 1, false));
  }
}

// ==================================================================
// Kernel C: single-wave sequential greedy scan + output emission.
// Lane owns 3 contiguous words -> one b96 load per kept row; words
// below the diagonal (never written by kernel B) are masked to zero;
// keep-bit broadcast via __shfl; rows prefetched (global_prefetch_b8).
// ==================================================================
__global__ __launch_bounds__(32)
void nms_scan_out(const float* __restrict__ ws, const unsigned* __restrict__ mask32,
                  float* __restrict__ outF) {
  __shared__ int keepList[MAX_DETS];
  const int lane = threadIdx.x;
  const int nvalid = ((const int*)ws)[WNV];
  const int w0 = 3 * lane;                 // lane owns words w0, w0+1, w0+2

  unsigned rem0 = 0, rem1 = 0, rem2 = 0;
  int cnt = 0;

  for (int i = 0; i < M_SEL; ++i) {
    if (lane < 6)
      __builtin_prefetch((const char*)mask32 + (size_t)(i + 24) * (NW32 * 4) + lane * 64, 0, 0);

    const int w = i >> 5;
    const int owner = w / 3;
    const int slot  = w - owner * 3;       // uniform per i
    unsigned myw = (slot == 0) ? rem0 : ((slot == 1) ? rem1 : rem2);
    unsigned wv = (unsigned)__shfl((int)myw, owner, 32);
    bool keep = (i < nvalid) && (((wv >> (i & 31)) & 1u) == 0u);
    if (keep) {
      if (lane == 0) keepList[cnt] = i;
      ++cnt;
      if (cnt >= MAX_DETS) break;
      v3u r = *(const v3u*)(mask32 + (size_t)i * NW32 + w0);  // one b96 load
      // diagonal masking: kernel B writes only tiles tj >= ti, so words
      // below i>>5 are stale; the word at i>>5 keeps only bits > (i&31).
      const unsigned dm = ~((2u << (i & 31)) - 1u);           // bits above i
      unsigned a = (w0     < w) ? 0u : ((w0     == w) ? (r.x & dm) : r.x);
      unsigned b = (w0 + 1 < w) ? 0u : ((w0 + 1 == w) ? (r.y & dm) : r.y);
      unsigned c = (w0 + 2 < w) ? 0u : ((w0 + 2 == w) ? (r.z & dm) : r.z);
      rem0 |= a;
      rem1 |= b;
      rem2 |= c;
    }
  }
  __syncthreads();

  int* outI = (int*)outF;
  for (int o = lane; o < MAX_DETS; o += 32) {
    bool vld = (o < cnt);
    float s = 0;
    v4f bb = { 0.0f, 0.0f, 0.0f, 0.0f };
    int ki = -1;
    if (vld) {
      int i = keepList[o];
      s  = ws[WS_ + i];
      bb = *(const v4f*)(ws + 4 * (size_t)i);                 // one b128 load
      ki = ((const int*)ws)[WORD_ + i];
    }
    float* r = outF + (size_t)o * 6;
    if (vld) { r[0] = 0.0f;  r[1] = s;    r[2] = bb.x; r[3] = bb.y; r[4] = bb.z; r[5] = bb.w; }
    else     { r[0] = -1.0f; r[1] = 0.0f; r[2] = 0.0f; r[3] = 0.0f; r[4] = 0.0f; r[5] = 0.0f; }
    outI[MAX_DETS * 6 + 1 + o] = ki;      // nms_keep_idx (int32 bits)
  }
  if (lane == 0) outI[MAX_DETS * 6] = cnt;  // kept_count (int32 bits)
}

// ==================================================================
extern "C" void kernel_launch(void* const* d_in, const int* in_sizes, int n_in,
                              void* d_out, int out_size, void* d_ws, size_t ws_size,
                              hipStream_t stream) {
  (void)in_sizes; (void)n_in; (void)out_size; (void)ws_size;
  const float* yolo = (const float*)d_in[0];
  const int*   bidx = (const int*)d_in[1];
  float* ws = (float*)d_ws;
  unsigned short* mask16 = (unsigned short*)((char*)d_ws + MASK_BYTE_OFF);
  const unsigned* mask32 = (const unsigned*)((char*)d_ws + MASK_BYTE_OFF);

  nms_gather_sort<<<dim3(1),  dim3(1024), 0, stream>>>(yolo, bidx, ws);
  nms_mask       <<<dim3(24), dim3(128),  0, stream>>>(ws, mask16);
  nms_scan_out   <<<dim3(1),  dim3(32),   0, stream>>>(ws, mask32, (float*)d_out);
}